// CompGCNLayer2_12180527251910
// MI455X (gfx1250) — compile-verified
//
#include <hip/hip_runtime.h>

typedef __attribute__((ext_vector_type(2))) float v2f;
typedef __attribute__((ext_vector_type(8))) float v8f;

#define N_NODES 50000
#define N_EDGES 600000
#define DIM     128
#define S2      144   // pair-row stride in float2 units; 2*144 mod 64 == 32 -> conflict-free b64

// ---------------------------------------------------------------------------
// Kernel 1: zero the scatter accumulator (pre_agg) in workspace.
// ---------------------------------------------------------------------------
__global__ __launch_bounds__(256) void zero_kernel(float* __restrict__ p, int n4) {
    int i = blockIdx.x * blockDim.x + threadIdx.x;
    if (i < n4) {
        ((float4*)p)[i] = make_float4(0.f, 0.f, 0.f, 0.f);
    }
}

// ---------------------------------------------------------------------------
// Kernel 2: per-edge scatter: pre_agg[dst,:] += h[src,:] + rel_emb[type,:]
// One wave32 per edge; each lane owns 4 contiguous floats (32*4 = 128 = DIM).
// h (25.6MB) and rel_emb (256KB) are L2-resident; atomics resolve in L2.
// ---------------------------------------------------------------------------
__global__ __launch_bounds__(256) void scatter_kernel(
    const float* __restrict__ h, const float* __restrict__ rel,
    const int* __restrict__ esrc, const int* __restrict__ edst,
    const int* __restrict__ etype, float* __restrict__ pre) {
    int gid = blockIdx.x * blockDim.x + threadIdx.x;
    int e = gid >> 5;
    if (e >= N_EDGES) return;
    int lane = threadIdx.x & 31;

    int src = esrc[e];
    int dst = edst[e];
    int t   = etype[e];

    const float4 hv = *(const float4*)(h   + (size_t)src * DIM + lane * 4);
    const float4 rv = *(const float4*)(rel + (size_t)t   * DIM + lane * 4);

    float* p = pre + (size_t)dst * DIM + lane * 4;
    __hip_atomic_fetch_add(p + 0, hv.x + rv.x, __ATOMIC_RELAXED, __HIP_MEMORY_SCOPE_AGENT);
    __hip_atomic_fetch_add(p + 1, hv.y + rv.y, __ATOMIC_RELAXED, __HIP_MEMORY_SCOPE_AGENT);
    __hip_atomic_fetch_add(p + 2, hv.z + rv.z, __ATOMIC_RELAXED, __HIP_MEMORY_SCOPE_AGENT);
    __hip_atomic_fetch_add(p + 3, hv.w + rv.w, __ATOMIC_RELAXED, __HIP_MEMORY_SCOPE_AGENT);
}

// ---------------------------------------------------------------------------
// Kernel 3: fused GEMM + epilogue, full fp32 via V_WMMA_F32_16X16X4_F32:
//   out = relu( (pre_agg * norm_row) @ W_neighbor + h @ loop_weight )
// One wave per 16-row x 128-col output block; 8 waves per workgroup share the
// B operand staged in LDS. B is staged as interleaved even/odd ROW PAIRS:
//   sP[r/2][n] = float2{ B[r][n], B[r+1][n] }
// so each WMMA B operand is a single aligned ds_load_b64 (no register
// reshuffles), and the pair-row stride S2=144 makes the two half-waves hit
// disjoint 32-bank halves (conflict-free).
//
// ISA VGPR layouts (wave32), lo = lane&15, hi = lane>>4:
//   A 16x4 : a[v]    = A[lo][k + v + 2*hi]            (contiguous float2 load)
//   B 4x16 : b[v]    = B[k + v + 2*hi][n0 + lo]  ==  sP[k/2 + hi][n0 + lo]
//   C 16x16: acc[r]  = C[m0 + r + 8*hi][n0 + lo]
// ---------------------------------------------------------------------------
__global__ __launch_bounds__(256) void gemm_kernel(
    const float* __restrict__ pre, const float* __restrict__ h,
    const float* __restrict__ norm,
    const float* __restrict__ Wn, const float* __restrict__ Lw,
    float* __restrict__ out) {
    __shared__ v2f sP[(DIM / 2) * S2];   // 64*144*8B = 72KB; 4 workgroups/WGP

    const int tid  = threadIdx.x;
    const int wave = (blockIdx.x * blockDim.x + tid) >> 5;
    const int ntiles = N_NODES / 16;     // 3125
    const bool active = (wave < ntiles); // wave-uniform; EXEC all-1s inside

    const int lane = tid & 31;
    const int lo = lane & 15;
    const int hi = lane >> 4;
    const int m0 = wave * 16;
    const int row = m0 + lo;

    // ---- stage W_neighbor into LDS as even/odd row pairs ----
    for (int i = tid; i < (DIM / 2) * DIM; i += 256) {
        int p = i >> 7;       // pair-row 0..63
        int n = i & (DIM - 1);
        v2f v;
        v.x = Wn[(2 * p) * DIM + n];
        v.y = Wn[(2 * p + 1) * DIM + n];
        sP[p * S2 + n] = v;
    }
    __syncthreads();

    v8f acc[8];
#pragma unroll
    for (int t = 0; t < 8; ++t) acc[t] = (v8f)(0.f);

    if (active) {
        const float nrm = norm[row];
        const float* Arow = pre + (size_t)row * DIM;
#pragma unroll 4
        for (int k = 0; k < DIM; k += 4) {
            v2f a = *(const v2f*)(Arow + k + 2 * hi);
            a.x *= nrm;
            a.y *= nrm;
            const v2f* bb = &sP[(k / 2 + hi) * S2 + lo];
#pragma unroll
            for (int t = 0; t < 8; ++t) {
                v2f b = bb[t * 16];
                acc[t] = __builtin_amdgcn_wmma_f32_16x16x4_f32(
                    false, a, false, b, (short)0, acc[t], false, false);
            }
        }
    }
    __syncthreads();

    // ---- stage loop_weight into LDS as even/odd row pairs ----
    for (int i = tid; i < (DIM / 2) * DIM; i += 256) {
        int p = i >> 7;
        int n = i & (DIM - 1);
        v2f v;
        v.x = Lw[(2 * p) * DIM + n];
        v.y = Lw[(2 * p + 1) * DIM + n];
        sP[p * S2 + n] = v;
    }
    __syncthreads();

    if (active) {
        const float* Hrow = h + (size_t)row * DIM;
#pragma unroll 4
        for (int k = 0; k < DIM; k += 4) {
            v2f a = *(const v2f*)(Hrow + k + 2 * hi);
            const v2f* bb = &sP[(k / 2 + hi) * S2 + lo];
#pragma unroll
            for (int t = 0; t < 8; ++t) {
                v2f b = bb[t * 16];
                acc[t] = __builtin_amdgcn_wmma_f32_16x16x4_f32(
                    false, a, false, b, (short)0, acc[t], false, false);
            }
        }

        // Epilogue: ReLU + store, per the 16x16 f32 C/D layout.
#pragma unroll
        for (int t = 0; t < 8; ++t) {
#pragma unroll
            for (int r = 0; r < 8; ++r) {
                float v = acc[t][r];
                v = v > 0.f ? v : 0.f;
                out[(size_t)(m0 + r + 8 * hi) * DIM + t * 16 + lo] = v;
            }
        }
    }
}

// ---------------------------------------------------------------------------
extern "C" void kernel_launch(void* const* d_in, const int* in_sizes, int n_in,
                              void* d_out, int out_size, void* d_ws, size_t ws_size,
                              hipStream_t stream) {
    const float* h    = (const float*)d_in[0];
    const float* norm = (const float*)d_in[1];
    const float* rel  = (const float*)d_in[2];
    const float* Wn   = (const float*)d_in[3];
    const float* Lw   = (const float*)d_in[4];
    const int* esrc   = (const int*)d_in[5];
    const int* edst   = (const int*)d_in[6];
    const int* etype  = (const int*)d_in[7];
    float* out = (float*)d_out;
    float* pre = (float*)d_ws;  // N_NODES * DIM floats = 25.6 MB scratch

    // 1) zero the accumulator
    int n4 = (N_NODES * DIM) / 4;
    zero_kernel<<<(n4 + 255) / 256, 256, 0, stream>>>(pre, n4);

    // 2) edge scatter (one wave per edge)
    long long sthreads = (long long)N_EDGES * 32;
    int sblocks = (int)((sthreads + 255) / 256);
    scatter_kernel<<<sblocks, 256, 0, stream>>>(h, rel, esrc, edst, etype, pre);

    // 3) fused WMMA GEMM + ReLU (one wave per 16-row block, LDS-staged B pairs)
    long long gthreads = (long long)(N_NODES / 16) * 32;
    int gblocks = (int)((gthreads + 255) / 256);
    gemm_kernel<<<gblocks, 256, 0, stream>>>(pre, h, norm, Wn, Lw, out);
}